// GMM_49400713839280
// MI455X (gfx1250) — compile-verified
//
#include <hip/hip_runtime.h>
#include <math.h>

typedef __attribute__((ext_vector_type(2))) float v2f;
typedef __attribute__((ext_vector_type(8))) float v8f;

#define LOG2E 1.44269504088896340736f

// One wave32 per 16-point tile, transposed orientation:
//   C[k][point] = W1 @ (X*X)aug + W2 @ Xaug   via v_wmma_f32_16x16x4_f32
// A (16x4) = constant weight rows (k = M, rows 8..15 zero), B (4x16) = data
// columns (point = N). C VGPR i on lanes 0..15 = (k=i, point=lane): the
// normalization over k is an in-lane sum -> no cross-lane shuffles.
__global__ __launch_bounds__(256) void gmm_posterior_wmma(
    const float* __restrict__ X,      // (n,16)
    const float* __restrict__ phi,    // (8,)
    const float* __restrict__ mu,     // (8,16)
    const float* __restrict__ sigma,  // (8,16)
    float* __restrict__ out,          // (8,n)
    int n)
{
    const int lane = threadIdx.x & 31;
    const int l15  = lane & 15;        // A row (=k) ; B column (=point-in-tile)
    const int half = lane >> 4;        // selects K-dim sub-columns of the 4-wide step
    const int wavesPerBlock = blockDim.x >> 5;
    const int wave   = blockIdx.x * wavesPerBlock + (threadIdx.x >> 5);
    const int nWaves = gridDim.x * wavesPerBlock;
    const int nTiles = (n + 15) >> 4;

    // ---- per-wave constants: A fragments (weights) + log2-domain biases ----
    // A 16x4 layout (ISA): lane l15 holds row M=l15, element v = K = 2*half+v.
    // Step j covers dims 4j..4j+3:  w[j] element v = W[k=l15][d = 4j+2*half+v].
    v2f w1[4], w2[4];
    #pragma unroll
    for (int j = 0; j < 4; ++j) { w1[j].x = 0.f; w1[j].y = 0.f; w2[j].x = 0.f; w2[j].y = 0.f; }
    float lw2 = -__builtin_inff();
    if (l15 < 8) {
        const int k = l15;
        float slog2 = 0.f, cq = 0.f;
        #pragma unroll
        for (int d = 0; d < 16; ++d) {
            float s  = sigma[k * 16 + d];
            float m  = mu[k * 16 + d];
            float iv = 1.0f / s;
            slog2 += __builtin_amdgcn_logf(s);   // log2
            cq    += m * m * iv;
        }
        #pragma unroll
        for (int j = 0; j < 4; ++j) {
            const int d0 = 4 * j + 2 * half;
            float i0 = 1.0f / sigma[k * 16 + d0];
            float i1 = 1.0f / sigma[k * 16 + d0 + 1];
            w1[j].x = i0;
            w1[j].y = i1;
            w2[j].x = -2.0f * mu[k * 16 + d0]     * i0;
            w2[j].y = -2.0f * mu[k * 16 + d0 + 1] * i1;
        }
        // dens_k = 2^( lw2_k - 0.5*log2e*quad );  (2*pi)^16 cancels in posterior
        lw2 = __builtin_amdgcn_logf(phi[k]) - 0.5f * slog2 - 0.5f * cq * LOG2E;
    }
    // broadcast per-k bias so every lane holds all 8 (k = C VGPR index now)
    float lwb[8];
    #pragma unroll
    for (int i = 0; i < 8; ++i) lwb[i] = __shfl(lw2, i, 32);

    // ---- grid-stride loop over 16-point tiles (wave-uniform -> EXEC all-1s) ----
    for (int t = wave; t < nTiles; t += nWaves) {
        const int rbase = t << 4;
        int r = rbase + l15;
        if (r >= n) r = n - 1;                    // clamp tail-tile loads
        const float* rp = X + (size_t)r * 16 + 2 * half;

        if (t + nWaves < nTiles)                  // prefetch next tile's rows
            __builtin_prefetch(X + ((size_t)rbase + (size_t)nWaves * 16 + l15) * 16, 0, 3);

        // B fragments: element v = Xaug[d = 4j+2*half+v][point = l15]
        v2f a0 = *(const v2f*)(rp + 0);
        v2f a1 = *(const v2f*)(rp + 4);
        v2f a2 = *(const v2f*)(rp + 8);
        v2f a3 = *(const v2f*)(rp + 12);
        v2f q0 = a0 * a0, q1 = a1 * a1, q2 = a2 * a2, q3 = a3 * a3;

        v8f acc = {0.f, 0.f, 0.f, 0.f, 0.f, 0.f, 0.f, 0.f};
        acc = __builtin_amdgcn_wmma_f32_16x16x4_f32(false, w1[0], false, q0, (short)0, acc, false, false);
        acc = __builtin_amdgcn_wmma_f32_16x16x4_f32(false, w2[0], false, a0, (short)0, acc, false, false);
        acc = __builtin_amdgcn_wmma_f32_16x16x4_f32(false, w1[1], false, q1, (short)0, acc, false, false);
        acc = __builtin_amdgcn_wmma_f32_16x16x4_f32(false, w2[1], false, a1, (short)0, acc, false, false);
        acc = __builtin_amdgcn_wmma_f32_16x16x4_f32(false, w1[2], false, q2, (short)0, acc, false, false);
        acc = __builtin_amdgcn_wmma_f32_16x16x4_f32(false, w2[2], false, a2, (short)0, acc, false, false);
        acc = __builtin_amdgcn_wmma_f32_16x16x4_f32(false, w1[3], false, q3, (short)0, acc, false, false);
        acc = __builtin_amdgcn_wmma_f32_16x16x4_f32(false, w2[3], false, a3, (short)0, acc, false, false);

        // ---- epilogue: dens_k = exp2(-0.5*log2e*quad + lw_k), in-lane k-sum ----
        float dv[8];
        float s = 0.f;
        #pragma unroll
        for (int i = 0; i < 8; ++i) {
            dv[i] = __builtin_amdgcn_exp2f(fmaf(acc[i], -0.5f * LOG2E, lwb[i]));
            s += dv[i];
        }
        const float rs = __builtin_amdgcn_rcpf(s);

        // stores: VGPR i across lanes 0..15 = 16 consecutive points of out row k=i
        if (lane < 16 && (rbase + l15) < n) {
            float* dst = out + (size_t)rbase + (size_t)l15;
            #pragma unroll
            for (int i = 0; i < 8; ++i)
                dst[(size_t)i * (size_t)n] = dv[i] * rs;
        }
    }
}

extern "C" void kernel_launch(void* const* d_in, const int* in_sizes, int n_in,
                              void* d_out, int out_size, void* d_ws, size_t ws_size,
                              hipStream_t stream) {
    const float* X     = (const float*)d_in[0];
    const float* phi   = (const float*)d_in[1];
    const float* mu    = (const float*)d_in[2];
    const float* sigma = (const float*)d_in[3];
    float* out = (float*)d_out;

    const int n      = in_sizes[0] / 16;           // D = 16
    const int nTiles = (n + 15) / 16;
    const int wavesPerBlock = 8;                   // 256 threads = 8 wave32s
    int blocks = (nTiles + wavesPerBlock - 1) / wavesPerBlock;
    if (blocks > 2048) blocks = 2048;              // ~8 tiles/wave to amortize setup
    if (blocks < 1) blocks = 1;

    gmm_posterior_wmma<<<blocks, 256, 0, stream>>>(X, phi, mu, sigma, out, n);
}